// GAT_47150150975823
// MI455X (gfx1250) — compile-verified
//
#include <hip/hip_runtime.h>

typedef __attribute__((ext_vector_type(2))) float v2f;
typedef __attribute__((ext_vector_type(4))) float v4f;
typedef __attribute__((ext_vector_type(8))) float v8f;
typedef __attribute__((ext_vector_type(4))) int   v4i;

#define B_   4
#define N_   1024
#define FN_  128
#define FH_  128
#define FE_  16
#define FG_  128
#define ZIN_ 256
#define OUT_ 128
#define H_   8

#define EROWS   512          // (i,j) pairs per 16x32 tile
#define ESTRIDE 20           // dwords per staged edge row (80B, 16B aligned)

// ---- CDNA5 async global->LDS copy (double-buffered edge stream) ----------
// Signature (from clang diagnostic): (int4 AS1*, int4 AS3*, imm offset, imm cpol)
#if __has_builtin(__builtin_amdgcn_global_load_async_to_lds_b128)
#define ASYNC_COPY16(src, dst)                                                  \
    __builtin_amdgcn_global_load_async_to_lds_b128(                             \
        (__attribute__((address_space(1))) v4i*)(src),                          \
        (__attribute__((address_space(3))) v4i*)(dst), 0, 0)
#define HAVE_ASYNC_LDS 1
#else
#define ASYNC_COPY16(src, dst)                                                  \
    do { *(v4f*)(dst) = *(const v4f*)(src); } while (0)
#define HAVE_ASYNC_LDS 0
#endif

#if HAVE_ASYNC_LDS
#if __has_builtin(__builtin_amdgcn_s_wait_asynccnt)
#define WAIT_ASYNC0() __builtin_amdgcn_s_wait_asynccnt(0)
#else
#define WAIT_ASYNC0() asm volatile("s_wait_asynccnt 0x0" ::: "memory")
#endif
#else
#define WAIT_ASYNC0() do {} while (0)
#endif

// ---------------------------------------------------------------------------
// Kernel A: fused projections  z @ {m_w, skip_w, a1_w|a2_w}  via f32 WMMA.
// One block = one 16-row tile of z (B*N/16 = 256 blocks, 256 threads = 8 waves)
// ---------------------------------------------------------------------------
__global__ __launch_bounds__(256)
void gat_proj_kernel(const float* __restrict__ node, const float* __restrict__ hidden,
                     const float* __restrict__ m_w,  const float* __restrict__ m_b,
                     const float* __restrict__ skip_w, const float* __restrict__ skip_b,
                     const float* __restrict__ a1_w, const float* __restrict__ a1_b,
                     const float* __restrict__ a2_w, const float* __restrict__ a2_b,
                     float* __restrict__ V, float* __restrict__ SK,
                     float* __restrict__ AT1, float* __restrict__ AT2)
{
    __shared__ float zl[16 * 260];            // 16 rows x 256 cols, pad 260 (bank-safe)
    const int tile = blockIdx.x;
    const int g0   = tile * 16;
    const int t    = threadIdx.x;

    {   // cooperative staging of z tile (concat node|hidden)
        int r = t >> 4;
        int g = g0 + r;
        int b = g >> 10, n = g & 1023;
        const float* nf = node   + ((size_t)b * N_ + n) * FN_;
        const float* hf = hidden + ((size_t)b * N_ + n) * FH_;
        #pragma unroll
        for (int q = 0; q < 16; ++q) {
            int c = (t & 15) + q * 16;
            zl[r * 260 + c] = (c < FN_) ? nf[c] : hf[c - FN_];
        }
    }
    __syncthreads();

    const int lane = t & 31;
    const int wv   = t >> 5;                  // 0..7
    const int m16  = lane & 15;
    const int hi   = lane >> 4;

    for (int task = wv; task < 17; task += 8) {
        v8f acc;
        #pragma unroll
        for (int r = 0; r < 8; ++r) acc[r] = 0.f;

        if (task < 16) {                      // value tiles (0-7) / skip tiles (8-15)
            const float* W   = (task < 8) ? m_w : skip_w;
            const int   col0 = 16 * (task & 7);
            const float* Wc  = W + col0 + m16;
            for (int k0 = 0; k0 < ZIN_; k0 += 4) {
                int ka = k0 + 2 * hi;
                v2f a, b;
                a.x = zl[m16 * 260 + ka];
                a.y = zl[m16 * 260 + ka + 1];
                b.x = Wc[(size_t)ka * OUT_];
                b.y = Wc[(size_t)(ka + 1) * OUT_];
                acc = __builtin_amdgcn_wmma_f32_16x16x4_f32(false, a, false, b,
                                                            (short)0, acc, false, false);
            }
            const float bias = (task < 8) ? m_b[col0 + m16] : skip_b[col0 + m16];
            float* Obase = (task < 8) ? V : SK;
            #pragma unroll
            for (int r = 0; r < 8; ++r) {
                int m = r + 8 * hi;
                Obase[(size_t)(g0 + m) * OUT_ + col0 + m16] = acc[r] + bias;
            }
        } else {                              // packed attention tile: cols 0-7 a1, 8-15 a2
            const int c = m16;
            for (int k0 = 0; k0 < ZIN_; k0 += 4) {
                int ka = k0 + 2 * hi;
                v2f a, b;
                a.x = zl[m16 * 260 + ka];
                a.y = zl[m16 * 260 + ka + 1];
                b.x = (c < 8) ? a1_w[ka * H_ + c]       : a2_w[ka * H_ + (c - 8)];
                b.y = (c < 8) ? a1_w[(ka + 1) * H_ + c] : a2_w[(ka + 1) * H_ + (c - 8)];
                acc = __builtin_amdgcn_wmma_f32_16x16x4_f32(false, a, false, b,
                                                            (short)0, acc, false, false);
            }
            const float bias = (c < 8) ? a1_b[c] : a2_b[c - 8];
            #pragma unroll
            for (int r = 0; r < 8; ++r) {
                int m = r + 8 * hi;
                float vv = acc[r] + bias;
                if (c < 8) AT1[(size_t)(g0 + m) * H_ + c]       = vv;
                else       AT2[(size_t)(g0 + m) * H_ + (c - 8)] = vv;
            }
        }
    }
}

// ---------------------------------------------------------------------------
// Kernel B: att_g = graph_fts @ ag_w + ag_b   (tiny)
// ---------------------------------------------------------------------------
__global__ void gat_attg_kernel(const float* __restrict__ gf,
                                const float* __restrict__ ag_w,
                                const float* __restrict__ ag_b,
                                float* __restrict__ ATG)
{
    int t = threadIdx.x;
    if (t >= B_ * H_) return;
    int b = t >> 3, h = t & 7;
    float s = ag_b[h];
    for (int e = 0; e < FG_; ++e) s = fmaf(gf[b * FG_ + e], ag_w[e * H_ + h], s);
    ATG[t] = s;
}

// ---------------------------------------------------------------------------
// Kernel C: streaming fused attention. One block = (b, 16-row i-tile).
// Edge stream double-buffered into LDS with async global->LDS copies;
// online softmax over j; P*V via f32 WMMA (wave h owns head h).
// ---------------------------------------------------------------------------
__global__ __launch_bounds__(256)
void gat_attn_kernel(const float* __restrict__ edge, const int* __restrict__ adj,
                     const float* __restrict__ ae_w, const float* __restrict__ ae_b,
                     const float* __restrict__ V,    const float* __restrict__ SK,
                     const float* __restrict__ AT1,  const float* __restrict__ AT2,
                     const float* __restrict__ ATG,  float* __restrict__ out)
{
    extern __shared__ __align__(16) float ebuf[];   // [2][EROWS][ESTRIDE]
    __shared__ float sc[32 * 144];            // raw scores [j][i*9 + h], 9-padded
    __shared__ float aew[FE_][H_];
    __shared__ float a1s[16][H_];
    __shared__ float cb[H_];
    __shared__ float rowmax[H_][16];
    __shared__ float rowl[H_][16];
    __shared__ float rowsc[H_][16];

    const int b    = blockIdx.y;
    const int i0   = blockIdx.x * 16;
    const int t    = threadIdx.x;
    const int lane = t & 31;
    const int wv   = t >> 5;                  // == head for phase 2
    const int m16  = lane & 15;
    const int hi   = lane >> 4;

    // issue async copies for one 16i x 32j edge tile (thread t owns rows t, t+256)
    auto issue_tile = [&](int j0, int buf) {
        #pragma unroll
        for (int q = 0; q < 2; ++q) {
            int p = t + q * 256;
            int i = p >> 5, j = p & 31;
            const float* src = edge + (((size_t)b * N_ + i0 + i) * N_ + j0 + j) * FE_;
            float* dst = &ebuf[((size_t)buf * EROWS + p) * ESTRIDE];
            #pragma unroll
            for (int c = 0; c < 4; ++c)
                ASYNC_COPY16(src + 4 * c, dst + 4 * c);
        }
    };

    if (t < FE_ * H_) aew[t >> 3][t & 7] = ae_w[t];
    if (t < 16 * H_)  a1s[t >> 3][t & 7] = AT1[((size_t)b * N_ + i0 + (t >> 3)) * H_ + (t & 7)];
    if (t < H_)       cb[t] = ae_b[t] + ATG[b * H_ + t];
    if (lane < 16) { rowmax[wv][lane] = -3.0e38f; rowl[wv][lane] = 0.f; }

    issue_tile(0, 0);                          // prologue: prefetch first tile

    v8f O;
    #pragma unroll
    for (int r = 0; r < 8; ++r) O[r] = 0.f;
    __syncthreads();

    for (int j0 = 0; j0 < N_; j0 += 32) {
        const int buf = (j0 >> 5) & 1;
        WAIT_ASYNC0();                         // tile j0 now resident in ebuf[buf]
        if (j0 + 32 < N_) issue_tile(j0 + 32, buf ^ 1);   // stream next tile

        // ---- phase 1: scores for the 16i x 32j x 8h tile (edge from LDS) ----
        for (int p = t; p < 512; p += 256) {
            int i = p >> 5, j = p & 31;
            const float* er = &ebuf[((size_t)buf * EROWS + p) * ESTRIDE];
            float e0[FE_];
            #pragma unroll
            for (int e = 0; e < FE_; ++e) e0[e] = er[e];
            const float* a2 = AT2 + ((size_t)b * N_ + j0 + j) * H_;
            float bias = (float)(adj[((size_t)b * N_ + i0 + i) * N_ + j0 + j] - 1) * 1.0e9f;
            #pragma unroll
            for (int h = 0; h < H_; ++h) {
                float s = a1s[i][h] + cb[h] + a2[h];
                #pragma unroll
                for (int e = 0; e < FE_; ++e) s = fmaf(e0[e], aew[e][h], s);
                s = (s > 0.f) ? s : 0.01f * s;  // leaky_relu BEFORE adj bias
                sc[j * 144 + i * 9 + h] = s + bias;
            }
        }
        __syncthreads();

        // ---- phase 2: wave h -> online softmax + WMMA P*V for head h ----
        {
            const int h = wv;
            for (int i = 0; i < 16; ++i) {    // row-max update (lanes over j)
                float v = sc[lane * 144 + i * 9 + h];
                #pragma unroll
                for (int m = 16; m >= 1; m >>= 1) v = fmaxf(v, __shfl_xor(v, m, 32));
                float mo = rowmax[h][i];
                float mn = fmaxf(mo, v);
                if (lane == 0) { rowmax[h][i] = mn; rowsc[h][i] = __expf(mo - mn); }
            }
            #pragma unroll
            for (int r = 0; r < 8; ++r) O[r] *= rowsc[h][r + 8 * hi];  // rescale acc

            float mrow = rowmax[h][m16];
            float psum = 0.f;
            const float* Vb = V + ((size_t)b * N_ + j0) * OUT_ + 16 * h + m16;
            #pragma unroll
            for (int k0 = 0; k0 < 32; k0 += 4) {
                int jk = k0 + 2 * hi;
                v2f a, bb;
                a.x = __expf(sc[jk * 144 + m16 * 9 + h] - mrow);       // P fragment
                a.y = __expf(sc[(jk + 1) * 144 + m16 * 9 + h] - mrow);
                psum += a.x + a.y;                                     // fused row-sum
                bb.x = Vb[(size_t)jk * OUT_];
                bb.y = Vb[(size_t)(jk + 1) * OUT_];
                O = __builtin_amdgcn_wmma_f32_16x16x4_f32(false, a, false, bb,
                                                          (short)0, O, false, false);
            }
            psum += __shfl_xor(psum, 16, 32);  // lanes 0-15 now hold full row sums
            if (lane < 16) rowl[h][lane] = rowl[h][lane] * rowsc[h][lane] + psum;
        }
        __syncthreads();
    }

    // ---- epilogue: O/l + skip, ReLU ----
    {
        const int h = wv;
        #pragma unroll
        for (int r = 0; r < 8; ++r) {
            int m = r + 8 * hi;
            float linv = 1.0f / rowl[h][m];
            size_t idx = ((size_t)b * N_ + i0 + m) * OUT_ + 16 * h + m16;
            out[idx] = fmaxf(O[r] * linv + SK[idx], 0.f);
        }
    }
}

// ---------------------------------------------------------------------------
extern "C" void kernel_launch(void* const* d_in, const int* in_sizes, int n_in,
                              void* d_out, int out_size, void* d_ws, size_t ws_size,
                              hipStream_t stream)
{
    (void)in_sizes; (void)n_in; (void)out_size; (void)ws_size;
    const float* node   = (const float*)d_in[0];
    const float* edge   = (const float*)d_in[1];
    const float* graph  = (const float*)d_in[2];
    const int*   adj    = (const int*)  d_in[3];
    const float* hidden = (const float*)d_in[4];
    const float* m_w    = (const float*)d_in[5];
    const float* m_b    = (const float*)d_in[6];
    const float* skip_w = (const float*)d_in[7];
    const float* skip_b = (const float*)d_in[8];
    const float* a1_w   = (const float*)d_in[9];
    const float* a1_b   = (const float*)d_in[10];
    const float* a2_w   = (const float*)d_in[11];
    const float* a2_b   = (const float*)d_in[12];
    const float* ae_w   = (const float*)d_in[13];
    const float* ae_b   = (const float*)d_in[14];
    const float* ag_w   = (const float*)d_in[15];
    const float* ag_b   = (const float*)d_in[16];

    float* out = (float*)d_out;
    float* ws  = (float*)d_ws;
    float* V   = ws;                               // B*N*128
    float* SK  = V   + (size_t)B_ * N_ * OUT_;     // B*N*128
    float* AT1 = SK  + (size_t)B_ * N_ * OUT_;     // B*N*8
    float* AT2 = AT1 + (size_t)B_ * N_ * H_;       // B*N*8
    float* ATG = AT2 + (size_t)B_ * N_ * H_;       // B*8

    const size_t edge_lds_bytes = (size_t)2 * EROWS * ESTRIDE * sizeof(float); // 80 KB

    gat_proj_kernel<<<B_ * N_ / 16, 256, 0, stream>>>(node, hidden, m_w, m_b,
        skip_w, skip_b, a1_w, a1_b, a2_w, a2_b, V, SK, AT1, AT2);
    gat_attg_kernel<<<1, 32, 0, stream>>>(graph, ag_w, ag_b, ATG);
    gat_attn_kernel<<<dim3(N_ / 16, B_), 256, edge_lds_bytes, stream>>>(edge, adj,
        ae_w, ae_b, V, SK, AT1, AT2, ATG, out);
}